// SHOTDescriptor_24945170055733
// MI455X (gfx1250) — compile-verified
//
#include <hip/hip_runtime.h>
#include <hip/hip_bf16.h>
#include <math.h>

// Problem constants (match the reference)
#define BCLOUDS 4
#define NPTS    8192
#define KNB     5
#define LOCAL_BINS 10
#define TOTAL_BINS 80
#define QPB     128          // queries per block (8 waves x 16 queries)
#define IDXMASK 0x1FFFu      // 13 bits: NPTS-1

typedef __attribute__((ext_vector_type(2))) float v2f;
typedef __attribute__((ext_vector_type(8))) float v8f;

// ---------------------------------------------------------------------------
// Kernel 1: brute-force 5-NN per cloud using V_WMMA_F32_16X16X4_F32.
//   A (16x4, candidates): row m = (-2x, -2y, -2z, |c|^2)   <- stored in LDS
//   B (4x16, queries):    col n = ( x,   y,   z,  1   )
//   D[m][n] = |c_m|^2 - 2 c_m . q_n   ->  d2 = D + |q_n|^2 (per-lane scalar)
// C/D layout: lanes 0-15 hold N=lane, M=vgpr; lanes 16-31 hold N=lane-16,
// M=vgpr+8. Lane n and lane n+16 track top-5 for query n over disjoint
// candidate-row halves; merged with __shfl at the end (wave32).
// Top-5 state is packed u32 keys: dist bits (upper 19) | candidate idx (13).
// A-operand LDS loads are software-pipelined one chunk ahead so the dscnt
// wait overlaps the pack/guard arithmetic instead of stalling the WMMA.
// ---------------------------------------------------------------------------
__global__ __launch_bounds__(256) void shot_knn_kernel(
    const float* __restrict__ pts,   // [B*N,3]
    int* __restrict__ knn)           // [B*N,5] global point indices
{
    extern __shared__ float lds[];          // NPTS float4 = 128 KB
    float4* s4 = (float4*)lds;
    const float2* s2 = (const float2*)lds;

    const int blocksPerCloud = NPTS / QPB;  // 64
    const int b            = blockIdx.x / blocksPerCloud;
    const int blockInCloud = blockIdx.x % blocksPerCloud;
    const float* cloud = pts + (size_t)b * NPTS * 3;

    // Stage the whole cloud into LDS, pre-transformed for the A matrix:
    // (-2x, -2y, -2z, |p|^2)
    for (int i = threadIdx.x; i < NPTS; i += 256) {
        const float x = cloud[3 * i + 0];
        const float y = cloud[3 * i + 1];
        const float z = cloud[3 * i + 2];
        float4 v;
        v.x = -2.0f * x; v.y = -2.0f * y; v.z = -2.0f * z;
        v.w = x * x + y * y + z * z;
        s4[i] = v;
    }
    __syncthreads();

    const int wave = threadIdx.x >> 5;
    const int lane = threadIdx.x & 31;
    const int half = lane >> 4;             // 0: K rows {0,1}/M 0..7 ; 1: {2,3}/M 8..15
    const int l16  = lane & 15;

    // B matrix: 16 queries for this wave (constant across the chunk loop)
    const int q = blockInCloud * QPB + wave * 16 + l16;
    const float4 qv = s4[q];                // (-2x,-2y,-2z,|q|^2)
    const float sqq = qv.w;
    v2f bmat;
    bmat.x = half ? (-0.5f * qv.z) : (-0.5f * qv.x);  // K=0 / K=2
    bmat.y = half ? 1.0f           : (-0.5f * qv.y);  // K=1 / K=3

    // per-lane sorted top-5 packed keys (ascending)
    unsigned keys[KNB];
#pragma unroll
    for (int j = 0; j < KNB; ++j) keys[j] = 0xFFFFFFFFu;

    // per-lane A-operand LDS slot (float2 granularity): 2*(c) + half
    const int aslot = l16 * 2 + half;

    // prologue: load chunk 0's A operand
    float2 ab = s2[aslot];

    for (int chunk = 0; chunk < NPTS; chunk += 16) {
        v2f amat; amat.x = ab.x; amat.y = ab.y;

        // prefetch next chunk's A operand (wraps to slot 0 on last iter; unused)
        const int nxt = (chunk + 16) & (NPTS - 1);
        const float2 abn = s2[nxt * 2 + aslot];

        v8f acc = {0.f, 0.f, 0.f, 0.f, 0.f, 0.f, 0.f, 0.f};
        acc = __builtin_amdgcn_wmma_f32_16x16x4_f32(
            /*neg_a=*/false, amat, /*neg_b=*/false, bmat,
            /*c_mod=*/(short)0, acc, /*reuse_a=*/false, /*reuse_b=*/false);

        // pack 8 candidates: key = bits(max(d2,0))[31:13] | candidate_idx
        const unsigned baseIdx = (unsigned)(chunk + half * 8);
        unsigned k8[8];
#pragma unroll
        for (int r = 0; r < 8; ++r) {
            const float d2 = fmaxf(acc[r] + sqq, 0.0f);
            k8[r] = (__float_as_uint(d2) & ~IDXMASK) | (baseIdx + (unsigned)r);
        }
        // min-tree for skip guard
        unsigned kmin = min(min(min(k8[0], k8[1]), min(k8[2], k8[3])),
                            min(min(k8[4], k8[5]), min(k8[6], k8[7])));
        if (kmin < keys[KNB - 1]) {
#pragma unroll
            for (int r = 0; r < 8; ++r) {
                unsigned k = k8[r];
#pragma unroll
                for (int j = 0; j < KNB; ++j) {
                    const unsigned lo = min(k, keys[j]);
                    const unsigned hi = max(k, keys[j]);
                    keys[j] = lo; k = hi;
                }
            }
        }
        ab = abn;
    }

    // Merge the two candidate-half partitions: all lanes fetch partner state,
    // low half inserts partner's sorted 5 into its own sorted 5.
    unsigned ok[KNB];
#pragma unroll
    for (int j = 0; j < KNB; ++j)
        ok[j] = (unsigned)__shfl((int)keys[j], l16 + 16, 32);

    if (half == 0) {
#pragma unroll
        for (int j = 0; j < KNB; ++j) {
            unsigned k = ok[j];
#pragma unroll
            for (int t = 0; t < KNB; ++t) {
                const unsigned lo = min(k, keys[t]);
                const unsigned hi = max(k, keys[t]);
                keys[t] = lo; k = hi;
            }
        }
        const int gq = b * NPTS + q;
#pragma unroll
        for (int j = 0; j < KNB; ++j)
            knn[(size_t)gq * KNB + j] = b * NPTS + (int)(keys[j] & IDXMASK);
    }
}

// ---------------------------------------------------------------------------
// Analytic symmetric 3x3 eigendecomposition helpers (ascending eigenvalues).
// ---------------------------------------------------------------------------
__device__ inline void eigvec3(float a00, float a01, float a02,
                               float a11, float a12, float a22,
                               float lam, float* vx, float* vy, float* vz)
{
    const float r0x = a00 - lam, r0y = a01,       r0z = a02;
    const float r1x = a01,       r1y = a11 - lam, r1z = a12;
    const float r2x = a02,       r2y = a12,       r2z = a22 - lam;
    const float c0x = r0y * r1z - r0z * r1y, c0y = r0z * r1x - r0x * r1z, c0z = r0x * r1y - r0y * r1x;
    const float c1x = r0y * r2z - r0z * r2y, c1y = r0z * r2x - r0x * r2z, c1z = r0x * r2y - r0y * r2x;
    const float c2x = r1y * r2z - r1z * r2y, c2y = r1z * r2x - r1x * r2z, c2z = r1x * r2y - r1y * r2x;
    const float n0 = c0x * c0x + c0y * c0y + c0z * c0z;
    const float n1 = c1x * c1x + c1y * c1y + c1z * c1z;
    const float n2 = c2x * c2x + c2y * c2y + c2z * c2z;
    float bx = c0x, by = c0y, bz = c0z, bn = n0;
    if (n1 > bn) { bx = c1x; by = c1y; bz = c1z; bn = n1; }
    if (n2 > bn) { bx = c2x; by = c2y; bz = c2z; bn = n2; }
    if (bn < 1e-30f) { bx = 1.0f; by = 0.0f; bz = 0.0f; bn = 1.0f; }
    const float inv = rsqrtf(bn);
    *vx = bx * inv; *vy = by * inv; *vz = bz * inv;
}

// ---------------------------------------------------------------------------
// Kernel 2: per-point local reference frame (eigenvectors of 5-NN covariance).
// Stores [v0 | v1 | v2] (ascending eigenvalues) as 9 floats per point.
// v0 = normal (smallest eigenvalue).
// ---------------------------------------------------------------------------
__global__ __launch_bounds__(256) void shot_lrf_kernel(
    const float* __restrict__ pts,
    const int* __restrict__ knn,
    float* __restrict__ lrf,
    int total)
{
    const int i = blockIdx.x * blockDim.x + threadIdx.x;
    if (i >= total) return;

    float nx[KNB], ny[KNB], nz[KNB];
    float mx = 0.f, my = 0.f, mz = 0.f;
#pragma unroll
    for (int j = 0; j < KNB; ++j) {
        const int g = knn[(size_t)i * KNB + j];
        nx[j] = pts[3 * g + 0]; ny[j] = pts[3 * g + 1]; nz[j] = pts[3 * g + 2];
        mx += nx[j]; my += ny[j]; mz += nz[j];
    }
    const float ik = 1.0f / (float)KNB;
    mx *= ik; my *= ik; mz *= ik;

    float a00 = 0.f, a01 = 0.f, a02 = 0.f, a11 = 0.f, a12 = 0.f, a22 = 0.f;
#pragma unroll
    for (int j = 0; j < KNB; ++j) {
        const float dx = nx[j] - mx, dy = ny[j] - my, dz = nz[j] - mz;
        a00 += dx * dx; a01 += dx * dy; a02 += dx * dz;
        a11 += dy * dy; a12 += dy * dz; a22 += dz * dz;
    }
    a00 *= ik; a01 *= ik; a02 *= ik; a11 *= ik; a12 *= ik; a22 *= ik;

    float V[9];
    const float p1 = a01 * a01 + a02 * a02 + a12 * a12;
    const float qm = (a00 + a11 + a22) * (1.0f / 3.0f);
    const float b00 = a00 - qm, b11 = a11 - qm, b22 = a22 - qm;
    const float p2 = b00 * b00 + b11 * b11 + b22 * b22 + 2.0f * p1;
    if (p2 < 1e-24f) {
        V[0] = 1.f; V[1] = 0.f; V[2] = 0.f;
        V[3] = 0.f; V[4] = 1.f; V[5] = 0.f;
        V[6] = 0.f; V[7] = 0.f; V[8] = 1.f;
    } else {
        const float p  = sqrtf(p2 * (1.0f / 6.0f));
        const float ip = 1.0f / p;
        const float c00 = b00 * ip, c01 = a01 * ip, c02 = a02 * ip;
        const float c11 = b11 * ip, c12 = a12 * ip, c22 = b22 * ip;
        const float detB = c00 * (c11 * c22 - c12 * c12)
                         - c01 * (c01 * c22 - c12 * c02)
                         + c02 * (c01 * c12 - c11 * c02);
        const float r = fminf(1.0f, fmaxf(-1.0f, 0.5f * detB));
        const float phi = acosf(r) * (1.0f / 3.0f);
        const float l2 = qm + 2.0f * p * cosf(phi);                       // largest
        const float l0 = qm + 2.0f * p * cosf(phi + 2.0943951023931953f); // smallest
        eigvec3(a00, a01, a02, a11, a12, a22, l0, &V[0], &V[1], &V[2]);
        eigvec3(a00, a01, a02, a11, a12, a22, l2, &V[6], &V[7], &V[8]);
        // middle eigenvector: orthogonal complement
        float vx = V[7] * V[2] - V[8] * V[1];
        float vy = V[8] * V[0] - V[6] * V[2];
        float vz = V[6] * V[1] - V[7] * V[0];
        const float nn = vx * vx + vy * vy + vz * vz;
        const float inv = rsqrtf(fmaxf(nn, 1e-30f));
        V[3] = vx * inv; V[4] = vy * inv; V[5] = vz * inv;
    }
#pragma unroll
    for (int j = 0; j < 9; ++j) lrf[(size_t)i * 9 + j] = V[j];
}

// ---------------------------------------------------------------------------
// Kernel 3: per-point 80-bin SHOT-like descriptor.
// ---------------------------------------------------------------------------
__global__ __launch_bounds__(256) void shot_desc_kernel(
    const float* __restrict__ pts,
    const int* __restrict__ knn,
    const float* __restrict__ lrf,
    float* __restrict__ out,
    int total)
{
    const int i = blockIdx.x * blockDim.x + threadIdx.x;
    if (i >= total) return;

    float V[9];
#pragma unroll
    for (int j = 0; j < 9; ++j) V[j] = lrf[(size_t)i * 9 + j];
    const float n0x = V[0], n0y = V[1], n0z = V[2];    // normal = v0

    int bins[KNB];
#pragma unroll
    for (int j = 0; j < KNB; ++j) {
        const int g = knn[(size_t)i * KNB + j];
        const float px = pts[3 * g + 0], py = pts[3 * g + 1], pz = pts[3 * g + 2];
        // project raw neighbor coords onto LRF columns
        const float p0 = px * V[0] + py * V[1] + pz * V[2];
        const float p1 = px * V[3] + py * V[4] + pz * V[5];
        const float p2 = px * V[6] + py * V[7] + pz * V[8];
        const int sid = ((p0 >= 0.f) ? 4 : 0) | ((p1 >= 0.f) ? 2 : 0) | ((p2 >= 0.f) ? 1 : 0);
        // neighbor normal = its v0
        const float mx = lrf[(size_t)g * 9 + 0];
        const float my = lrf[(size_t)g * 9 + 1];
        const float mz = lrf[(size_t)g * 9 + 2];
        const float cosv = n0x * mx + n0y * my + n0z * mz;
        float nid = floorf((float)LOCAL_BINS * (cosv + 1.0f) * 0.5f);
        nid = fminf(fmaxf(nid, 0.0f), (float)(LOCAL_BINS - 1));
        bins[j] = sid * LOCAL_BINS + (int)nid;
    }

    float* row = out + (size_t)i * TOTAL_BINS;
    for (int bbin = 0; bbin < TOTAL_BINS; ++bbin) {
        float c = 0.f;
#pragma unroll
        for (int j = 0; j < KNB; ++j) c += (bins[j] == bbin) ? 1.0f : 0.0f;
        row[bbin] = c;
    }
}

// ---------------------------------------------------------------------------
extern "C" void kernel_launch(void* const* d_in, const int* in_sizes, int n_in,
                              void* d_out, int out_size, void* d_ws, size_t ws_size,
                              hipStream_t stream)
{
    (void)in_sizes; (void)n_in; (void)out_size; (void)ws_size;
    const float* points = (const float*)d_in[0];   // [B*N, 3] f32
    // d_in[1] = batch ids (unused: equal-size sorted clouds)

    const int total = BCLOUDS * NPTS;              // 32768

    // workspace: knn [total*5] int, lrf [total*9] float  (~1.8 MB)
    int*   knn = (int*)d_ws;
    float* lrf = (float*)((char*)d_ws + (size_t)total * KNB * sizeof(int));

    // Kernel 1: 4 clouds x 64 blocks, 256 threads, 128 KB dynamic LDS
    shot_knn_kernel<<<BCLOUDS * (NPTS / QPB), 256, NPTS * sizeof(float4), stream>>>(
        points, knn);

    shot_lrf_kernel<<<(total + 255) / 256, 256, 0, stream>>>(points, knn, lrf, total);

    shot_desc_kernel<<<(total + 255) / 256, 256, 0, stream>>>(points, knn, lrf,
                                                              (float*)d_out, total);
}